// GCN_51007031607811
// MI455X (gfx1250) — compile-verified
//
#include <hip/hip_runtime.h>
#include <hip/hip_bf16.h>

typedef float v2f __attribute__((ext_vector_type(2)));
typedef float v8f __attribute__((ext_vector_type(8)));

#define N_NODE   200
#define NN_ELEM  40000          // 200*200
#define K_KEEP   12000          // 30% kept
#define HID      128
#define STR      132            // padded LDS row stride (floats) -> conflict-free row gathers
#define THREADS  256
#define ROW_TILES 13            // ceil(200/16) -> 208 padded rows (legal store targets)
#define COL_TILES 8             // 128/16
#define N_TILES  (ROW_TILES*COL_TILES)

// ---- LDS layout (bytes). bufM/bufH alias the |x|-bits staging area (phase A/B only).
#define OFF_M     0
#define OFF_H     (208*STR*4)                 // 109824
#define OFF_HIST  (2*208*STR*4)               // 219648 (2048 u32; reused for pooling)
#define OFF_MASK  (OFF_HIST + 8192)           // 227840 (200*8 u32)
#define OFF_DEG   (OFF_MASK + 6400)           // 234240 (200 u32)
#define OFF_DINV  (OFF_DEG + 800)             // 235040 (208 f32, zero-padded tail)
#define OFF_SCAL  (OFF_DINV + 832)            // 235872 (small scalars)
#define LDS_BYTES (OFF_SCAL + 64)             // 235936 total (< 320KB per WGP)

__device__ __forceinline__ v8f wmma4(v2f a, v2f b, v8f c) {
  // D = A(16x4,f32) * B(4x16,f32) + C(16x16,f32)   -> v_wmma_f32_16x16x4_f32
  return __builtin_amdgcn_wmma_f32_16x16x4_f32(false, a, false, b, (short)0, c,
                                               false, false);
}

extern "C" __global__ __launch_bounds__(THREADS)
void gcn_topk_fused(const float* __restrict__ x,
                    const float* __restrict__ W1, const float* __restrict__ b1,
                    const float* __restrict__ W2, const float* __restrict__ b2,
                    const float* __restrict__ Wout, const float* __restrict__ bout,
                    float* __restrict__ out)
{
  extern __shared__ char smem[];
  float*    bufM  = (float*)(smem + OFF_M);     // [208][STR] scaled features M
  float*    bufH  = (float*)(smem + OFF_H);     // [208][STR] hidden H
  unsigned* sAbs  = (unsigned*)(smem + OFF_M);  // [40000] |x| bit patterns (aliases bufM/H)
  unsigned* hist  = (unsigned*)(smem + OFF_HIST);
  unsigned* maskT = (unsigned*)(smem + OFF_MASK); // maskT[j*8+w] bits over source i
  unsigned* degc  = (unsigned*)(smem + OFF_DEG);
  float*    dinv  = (float*)(smem + OFF_DINV);    // [208], rows 200..207 = 0
  unsigned* scal  = (unsigned*)(smem + OFF_SCAL);
  float*    ppart = (float*)(smem + OFF_HIST);    // [256] pooling partials (reuse hist)
  float*    pooled= (float*)(smem + OFF_HIST) + 256; // [128]

  const int tid = threadIdx.x;
  const int b   = blockIdx.x;
  const float* xb = x + (size_t)b * NN_ELEM;

  // ======== Phase A: exact top-K threshold via 3-level radix histogram select ========
  for (int i = tid; i < 2048; i += THREADS) hist[i] = 0;
  __syncthreads();
  for (int i = tid; i < NN_ELEM; i += THREADS) {
    unsigned u = __float_as_uint(xb[i]) & 0x7fffffffu;
    sAbs[i] = u;
    atomicAdd(&hist[u >> 21], 1u);                       // bits [30:21], 1024 bins
  }
  __syncthreads();
  if (tid == 0) {
    unsigned cum = 0, sel = 0, R = K_KEEP;
    for (int bin = 1023; bin >= 0; --bin) {
      unsigned h = hist[bin];
      if (cum + h >= (unsigned)K_KEEP) { sel = (unsigned)bin; R = K_KEEP - cum; break; }
      cum += h;
    }
    scal[0] = sel; scal[1] = R;
  }
  __syncthreads();
  const unsigned sel1 = scal[0], R1 = scal[1];
  for (int i = tid; i < 2048; i += THREADS) hist[i] = 0;
  __syncthreads();
  for (int i = tid; i < NN_ELEM; i += THREADS) {
    unsigned u = sAbs[i];
    if ((u >> 21) == sel1) atomicAdd(&hist[(u >> 11) & 1023u], 1u);  // bits [20:11]
  }
  __syncthreads();
  if (tid == 0) {
    unsigned cum = 0, sel = 0, R = R1;
    for (int bin = 1023; bin >= 0; --bin) {
      unsigned h = hist[bin];
      if (cum + h >= R1) { sel = (unsigned)bin; R = R1 - cum; break; }
      cum += h;
    }
    scal[0] = sel; scal[1] = R;
  }
  __syncthreads();
  const unsigned sel2 = scal[0], R2 = scal[1];
  const unsigned pfx  = (sel1 << 10) | sel2;
  for (int i = tid; i < 2048; i += THREADS) hist[i] = 0;
  __syncthreads();
  for (int i = tid; i < NN_ELEM; i += THREADS) {
    unsigned u = sAbs[i];
    if ((u >> 11) == pfx) atomicAdd(&hist[u & 0x7ffu], 1u);          // bits [10:0]
  }
  __syncthreads();
  if (tid == 0) {
    unsigned cum = 0, sel = 0;
    for (int bin = 2047; bin >= 0; --bin) {
      unsigned h = hist[bin];
      if (cum + h >= R2) { sel = (unsigned)bin; break; }
      cum += h;
    }
    scal[2] = (sel1 << 21) | (sel2 << 11) | sel;         // exact K-th largest |x| bits
  }
  __syncthreads();
  const unsigned T = scal[2];

  // ======== Phase B: packed mask^T bitmap + degree -> dinv ========
  for (int j = tid; j < N_NODE; j += THREADS) { degc[j] = 0; maskT[j*8+7] = 0; }
  __syncthreads();
  for (int p = tid; p < N_NODE * 7; p += THREADS) {
    const int j = p / 7, w = p % 7;
    const int ibase = w * 32;
    const int imax  = (N_NODE - ibase) < 32 ? (N_NODE - ibase) : 32;
    unsigned bits = 0;
    for (int r = 0; r < imax; ++r) {
      unsigned u = sAbs[(ibase + r) * N_NODE + j];       // mask[i][j]
      bits |= (u >= T ? 1u : 0u) << r;
    }
    maskT[j * 8 + w] = bits;
    atomicAdd(&degc[j], (unsigned)__popc(bits));
  }
  __syncthreads();
  for (int j = tid; j < 208; j += THREADS)
    dinv[j] = (j < N_NODE) ? rsqrtf((float)(degc[j] + 1u)) : 0.0f;
  __syncthreads();                                        // sAbs dead after this point

  const int lane = tid & 31;
  const int wid  = tid >> 5;
  const int l    = lane & 15;
  const int hf   = lane >> 4;                             // lane half for f32 WMMA frags

  // ======== GEMM1: M = dinv * (X @ W1), X from global (L2-hot) ========
  // Padding rows (200..207) read clamped row 199 and write padding rows -> never read.
  for (int t = wid; t < N_TILES; t += 8) {
    const int m0 = (t >> 3) * 16, n0 = (t & 7) * 16;
    const int m  = m0 + l;
    const int mr = m < N_NODE ? m : N_NODE - 1;
    const float* arow = xb + mr * N_NODE;
    v8f c = {0.f,0.f,0.f,0.f,0.f,0.f,0.f,0.f};
    for (int k0 = 0; k0 < N_NODE; k0 += 4) {
      const int ka = k0 + 2 * hf;
      v2f a;  a.x = arow[ka];               a.y = arow[ka + 1];
      v2f bb; bb.x = W1[ka*HID + n0 + l];   bb.y = W1[(ka+1)*HID + n0 + l];
      c = wmma4(a, bb, c);
    }
#pragma unroll
    for (int v = 0; v < 8; ++v) {
      const int mm = m0 + v + 8 * hf;                    // unconditional padded store
      bufM[mm * STR + n0 + l] = dinv[mm] * c[v];
    }
  }
  __syncthreads();

  // ======== GEMM2: H1 = relu(dinv * (Ahat^T @ M) + b1), A synthesized from bitmask ====
  for (int t = wid; t < N_TILES; t += 8) {
    const int m0 = (t >> 3) * 16, n0 = (t & 7) * 16;
    const int j  = m0 + l;
    const int jr = j < N_NODE ? j : N_NODE - 1;
    const float bias = b1[n0 + l];
    v8f c = {0.f,0.f,0.f,0.f,0.f,0.f,0.f,0.f};
    for (int k0 = 0; k0 < N_NODE; k0 += 4) {
      const int ka = k0 + 2 * hf;                         // ka even -> ka,ka+1 same word
      const unsigned wv = maskT[jr * 8 + (ka >> 5)];
      v2f a;
      a.x = (float)((wv >> (ka & 31)) & 1u)        + (ka     == j ? 1.0f : 0.0f);
      a.y = (float)((wv >> ((ka & 31) + 1)) & 1u)  + (ka + 1 == j ? 1.0f : 0.0f);
      v2f bb; bb.x = bufM[ka*STR + n0 + l];  bb.y = bufM[(ka+1)*STR + n0 + l];
      c = wmma4(a, bb, c);
    }
#pragma unroll
    for (int v = 0; v < 8; ++v) {
      const int mm = m0 + v + 8 * hf;
      float hval = dinv[mm] * c[v] + bias;
      bufH[mm * STR + n0 + l] = hval > 0.f ? hval : 0.f;
    }
  }
  __syncthreads();

  // ======== GEMM3: M2 = dinv * (H1 @ W2), K = 128 ========
  for (int t = wid; t < N_TILES; t += 8) {
    const int m0 = (t >> 3) * 16, n0 = (t & 7) * 16;
    const int m  = m0 + l;
    const int mr = m < N_NODE ? m : N_NODE - 1;
    v8f c = {0.f,0.f,0.f,0.f,0.f,0.f,0.f,0.f};
    for (int k0 = 0; k0 < HID; k0 += 4) {
      const int ka = k0 + 2 * hf;
      v2f a;  a.x = bufH[mr*STR + ka];       a.y = bufH[mr*STR + ka + 1];
      v2f bb; bb.x = W2[ka*HID + n0 + l];    bb.y = W2[(ka+1)*HID + n0 + l];
      c = wmma4(a, bb, c);
    }
#pragma unroll
    for (int v = 0; v < 8; ++v) {
      const int mm = m0 + v + 8 * hf;
      bufM[mm * STR + n0 + l] = dinv[mm] * c[v];
    }
  }
  __syncthreads();

  // ======== GEMM4: H2 = relu(dinv * (Ahat^T @ M2) + b2) ========
  for (int t = wid; t < N_TILES; t += 8) {
    const int m0 = (t >> 3) * 16, n0 = (t & 7) * 16;
    const int j  = m0 + l;
    const int jr = j < N_NODE ? j : N_NODE - 1;
    const float bias = b2[n0 + l];
    v8f c = {0.f,0.f,0.f,0.f,0.f,0.f,0.f,0.f};
    for (int k0 = 0; k0 < N_NODE; k0 += 4) {
      const int ka = k0 + 2 * hf;
      const unsigned wv = maskT[jr * 8 + (ka >> 5)];
      v2f a;
      a.x = (float)((wv >> (ka & 31)) & 1u)        + (ka     == j ? 1.0f : 0.0f);
      a.y = (float)((wv >> ((ka & 31) + 1)) & 1u)  + (ka + 1 == j ? 1.0f : 0.0f);
      v2f bb; bb.x = bufM[ka*STR + n0 + l];  bb.y = bufM[(ka+1)*STR + n0 + l];
      c = wmma4(a, bb, c);
    }
#pragma unroll
    for (int v = 0; v < 8; ++v) {
      const int mm = m0 + v + 8 * hf;
      float hval = dinv[mm] * c[v] + bias;
      bufH[mm * STR + n0 + l] = hval > 0.f ? hval : 0.f;
    }
  }
  __syncthreads();

  // ======== Mean pool (all 256 threads: 2 row-segments x 128 features) ========
  {
    const int f   = tid & 127;
    const int seg = tid >> 7;                 // 0: rows 0..99, 1: rows 100..199
    float s = 0.f;
    for (int j = seg * 100; j < seg * 100 + 100; ++j) s += bufH[j * STR + f];
    ppart[tid] = s;
  }
  __syncthreads();
  if (tid < HID) pooled[tid] = (ppart[tid] + ppart[tid + 128]) * (1.0f / (float)N_NODE);
  __syncthreads();
  // ======== Output projection (128 -> 2) ========
  if (tid < 2) {
    float s = bout[tid];
    for (int f = 0; f < HID; ++f) s += pooled[f] * Wout[f * 2 + tid];
    out[b * 2 + tid] = s;
  }
}

extern "C" void kernel_launch(void* const* d_in, const int* in_sizes, int n_in,
                              void* d_out, int out_size, void* d_ws, size_t ws_size,
                              hipStream_t stream) {
  (void)n_in; (void)out_size; (void)d_ws; (void)ws_size;
  const float* x    = (const float*)d_in[0];
  // d_in[1] = adj (unused by the reference computation)
  const float* W1   = (const float*)d_in[2];
  const float* b1   = (const float*)d_in[3];
  const float* W2   = (const float*)d_in[4];
  const float* b2   = (const float*)d_in[5];
  const float* Wout = (const float*)d_in[6];
  const float* bout = (const float*)d_in[7];
  float* out = (float*)d_out;

  const int B = in_sizes[0] / NN_ELEM;   // 512
  gcn_topk_fused<<<dim3(B), dim3(THREADS), LDS_BYTES, stream>>>(
      x, W1, b1, W2, b2, Wout, bout, out);
}